// GraphEncoder_84413287236176
// MI455X (gfx1250) — compile-verified
//
#include <hip/hip_runtime.h>
#include <hip/hip_bf16.h>
#include <stddef.h>

typedef __attribute__((ext_vector_type(2))) float v2f;
typedef __attribute__((ext_vector_type(8))) float v8f;

#define KNN   16
#define NODES 2048
#define BATCH 8

// CDNA5 async global->LDS staging (ASYNCcnt), with safe fallback.
#if defined(__has_builtin)
#  if __has_builtin(__builtin_amdgcn_global_load_async_to_lds_b32) && \
      __has_builtin(__builtin_amdgcn_s_wait_asynccnt)
#    define USE_ASYNC_LDS 1
#  endif
#endif
#ifndef USE_ASYNC_LDS
#  define USE_ASYNC_LDS 0
#endif

typedef __attribute__((address_space(1))) int g_i32;  // global int
typedef __attribute__((address_space(3))) int l_i32;  // LDS int

// ---------------------------------------------------------------------------
// kNN: one block per 256-node chunk of one sample. xyz of the whole sample is
// staged in LDS (24 KB); each thread owns one query node and keeps a sorted
// top-16 (distance, index) entirely in VGPRs (static indexing only).
// ---------------------------------------------------------------------------
__global__ void knn_kernel(const float* __restrict__ coords,
                           int* __restrict__ idx_out) {
  __shared__ float sx[NODES], sy[NODES], sz[NODES];
  const int chunks = NODES / 256;                 // 8
  const int s      = blockIdx.x / chunks;         // sample
  const int chunk  = blockIdx.x % chunks;
  const float* base = coords + (size_t)s * NODES * 6;

  for (int j = threadIdx.x; j < NODES; j += 256) {
    sx[j] = base[(size_t)j * 6 + 0];
    sy[j] = base[(size_t)j * 6 + 1];
    sz[j] = base[(size_t)j * 6 + 2];
  }
  __syncthreads();

  const int i = chunk * 256 + threadIdx.x;
  const float xi = sx[i], yi = sy[i], zi = sz[i];

  float bd[KNN];
  int   bi[KNN];
#pragma unroll
  for (int k = 0; k < KNN; ++k) { bd[k] = 3.0e38f; bi[k] = 0; }

  for (int j = 0; j < NODES; ++j) {
    if (j == i) continue;
    const float dx = xi - sx[j];
    const float dy = yi - sy[j];
    const float dz = zi - sz[j];
    const float d  = dx * dx + dy * dy + dz * dz;
    if (d < bd[KNN - 1]) {
      // bubble-insert with static indices (stays in VGPRs)
      float dc = d; int ic = j;
#pragma unroll
      for (int k = 0; k < KNN; ++k) {
        const bool take = dc < bd[k];
        const float td = bd[k]; const int ti = bi[k];
        if (take) { bd[k] = dc; bi[k] = ic; dc = td; ic = ti; }
      }
    }
  }

  int* o = idx_out + ((size_t)s * NODES + i) * KNN;
#pragma unroll
  for (int k = 0; k < KNN; ++k) o[k] = bi[k];
}

// ---------------------------------------------------------------------------
// f32 WMMA GEMM: C[M x NDIM] = A[M x KDIM] * B[KDIM x NDIM] (+ bias).
// Block = 8 waves = 128 rows of M. B is staged zero-padded in LDS (via
// GLOBAL_LOAD_ASYNC_TO_LDS when available). Each wave register-caches its
// entire A strip (KPAD/4 v2f fragments, <= 64 VGPRs), then sweeps the N
// tiles with V_WMMA_F32_16X16X4_F32 reading only LDS — the inner loop has
// no global loads and no loadcnt waits.
// Fragment layout (wave32):
//   A 16x4 : lane<16 -> row=lane, K={kk,kk+1}; lane>=16 -> row=lane-16, K={kk+2,kk+3}
//   B 4x16 : lane<16 -> col=lane, K={kk,kk+1}; lane>=16 -> col=lane-16, K={kk+2,kk+3}
//   C 16x16: vgpr r -> row = r + 8*(lane>=16), col = lane&15
// ---------------------------------------------------------------------------
template <int KDIM, int NDIM>
__global__ void wmma_gemm(const float* __restrict__ A,
                          const float* __restrict__ B,
                          const float* __restrict__ bias,
                          float* __restrict__ C, int M) {
  constexpr int KPAD   = (KDIM + 3) & ~3;
  constexpr int KSTEPS = KPAD / 4;
  __shared__ float Bs[KPAD * NDIM];

#if USE_ASYNC_LDS
  {
    // Async DMA of the valid K x N block straight into LDS (no VGPR bounce),
    // plain DS stores for the zero-pad rows, then drain ASYNCcnt.
    for (int t = threadIdx.x; t < KDIM * NDIM; t += blockDim.x) {
      __builtin_amdgcn_global_load_async_to_lds_b32(
          (g_i32*)(B + t), (l_i32*)((l_i32*)Bs + t), 0, 0);
    }
    for (int t = KDIM * NDIM + threadIdx.x; t < KPAD * NDIM; t += blockDim.x)
      Bs[t] = 0.0f;
    __builtin_amdgcn_s_wait_asynccnt(0);
  }
#else
  for (int t = threadIdx.x; t < KPAD * NDIM; t += blockDim.x) {
    const int k = t / NDIM, n = t % NDIM;
    Bs[t] = (k < KDIM) ? B[(size_t)k * NDIM + n] : 0.0f;
  }
#endif
  __syncthreads();

  const int wave = threadIdx.x >> 5;
  const int lane = threadIdx.x & 31;
  const int half = lane >> 4;       // 0: K pair {0,1}, 1: K pair {2,3}
  const int l16  = lane & 15;
  const int m0   = (blockIdx.x * 8 + wave) * 16;
  if (m0 >= M) return;

  // Register-cache this wave's entire A strip.
  const float* Arow = A + (size_t)(m0 + l16) * KDIM;
  v2f afrag[KSTEPS];
#pragma unroll
  for (int ks = 0; ks < KSTEPS; ++ks) {
    const int c0 = 4 * ks + 2 * half;
    afrag[ks].x = (c0 + 0 < KDIM) ? Arow[c0 + 0] : 0.0f;
    afrag[ks].y = (c0 + 1 < KDIM) ? Arow[c0 + 1] : 0.0f;
  }

  for (int n0 = 0; n0 < NDIM; n0 += 16) {
    v8f acc = {};
#pragma unroll
    for (int ks = 0; ks < KSTEPS; ++ks) {
      const int c0 = 4 * ks + 2 * half;
      v2f b;
      b.x = Bs[(c0 + 0) * NDIM + n0 + l16];
      b.y = Bs[(c0 + 1) * NDIM + n0 + l16];
      acc = __builtin_amdgcn_wmma_f32_16x16x4_f32(
          /*neg_a=*/false, afrag[ks], /*neg_b=*/false, b,
          /*c_mod=*/(short)0, acc, /*reuse_a=*/false, /*reuse_b=*/false);
    }
    const float bv = bias ? bias[n0 + l16] : 0.0f;
    float* Crow = C + (size_t)(m0 + half * 8) * NDIM + n0 + l16;
#pragma unroll
    for (int r = 0; r < 8; ++r) Crow[(size_t)r * NDIM] = acc[r] + bv;
  }
}

// ---------------------------------------------------------------------------
// GCN aggregation: X[n,f] = relu((sum_k H[nbr[n,k],f] + H[n,f]) / 17 + b[f])
// One block per node, one thread per feature.
// ---------------------------------------------------------------------------
template <int F>
__global__ void agg_relu(const float* __restrict__ H,
                         const int* __restrict__ idx,
                         const float* __restrict__ bias,
                         float* __restrict__ X) {
  const int n = blockIdx.x;           // global node id (sample*2048 + i)
  const int f = threadIdx.x;
  const int base = (n >> 11) << 11;   // sample base row
  const int* nb = idx + (size_t)n * KNN;

  float sum = H[(size_t)n * F + f];
#pragma unroll
  for (int k = 0; k < KNN; ++k)
    sum += H[(size_t)(base + nb[k]) * F + f];

  X[(size_t)n * F + f] = fmaxf(sum * (1.0f / 17.0f) + bias[f], 0.0f);
}

// ---------------------------------------------------------------------------
extern "C" void kernel_launch(void* const* d_in, const int* in_sizes, int n_in,
                              void* d_out, int out_size, void* d_ws, size_t ws_size,
                              hipStream_t stream) {
  const float* coords = (const float*)d_in[0];  // [8,2048,6]
  const float* W1 = (const float*)d_in[1];      // [6,64]
  const float* b1 = (const float*)d_in[2];      // [64]
  const float* W2 = (const float*)d_in[3];      // [64,128]
  const float* b2 = (const float*)d_in[4];      // [128]
  const float* Wf = (const float*)d_in[5];      // [128,128]
  const float* bf = (const float*)d_in[6];      // [128]
  float* out = (float*)d_out;                   // [8,2048,128]

  const int Mtot = BATCH * NODES;               // 16384

  // Workspace layout (all L2-resident): ~25 MB
  char* w = (char*)d_ws;
  int*   idx = (int*)w;   w += (size_t)Mtot * KNN * sizeof(int);    // 1 MB
  float* H1  = (float*)w; w += (size_t)Mtot * 64  * sizeof(float);  // 4 MB
  float* X1  = (float*)w; w += (size_t)Mtot * 64  * sizeof(float);  // 4 MB
  float* H2  = (float*)w; w += (size_t)Mtot * 128 * sizeof(float);  // 8 MB
  float* X2  = (float*)w;                                           // 8 MB

  // 1) kNN graph per sample
  knn_kernel<<<BATCH * (NODES / 256), 256, 0, stream>>>(coords, idx);

  // 2) H1 = coords @ W1            (M=16384, K=6->pad 8, N=64)
  wmma_gemm<6, 64><<<Mtot / 128, 256, 0, stream>>>(coords, W1, nullptr, H1, Mtot);
  // 3) X1 = relu(agg(H1)/17 + b1)
  agg_relu<64><<<Mtot, 64, 0, stream>>>(H1, idx, b1, X1);

  // 4) H2 = X1 @ W2                (K=64, N=128)
  wmma_gemm<64, 128><<<Mtot / 128, 256, 0, stream>>>(X1, W2, nullptr, H2, Mtot);
  // 5) X2 = relu(agg(H2)/17 + b2)
  agg_relu<128><<<Mtot, 128, 0, stream>>>(H2, idx, b2, X2);

  // 6) out = X2 @ Wf + bf          (K=128, N=128)
  wmma_gemm<128, 128><<<Mtot / 128, 256, 0, stream>>>(X2, Wf, bf, out, Mtot);
}